// ModelNew_36215164240307
// MI455X (gfx1250) — compile-verified
//
#include <hip/hip_runtime.h>

// Mamba2 chunked SSD scan for MI455X (gfx1250, wave32).
// Bandwidth-bound (~200MB traffic vs ~4.3 GFLOP) -> keep fp32, use
// V_WMMA_F32_16X16X4_F32 for all four 64x64x64 matmuls per chunk.
// Each wave computes two 16x16 tiles sharing one B-fragment stream, with two
// independent accumulator chains interleaved to hide WMMA RAW latency.

#define BQ 2
#define SQ 4096
#define HQ 16
#define PQ 64
#define NQ 64
#define BL 64
#define CQ (SQ / BL)        // 64 chunks
#define LS 68               // LDS row stride (floats): rows 16B aligned, low bank conflicts
#define STATES_ELEMS ((size_t)BQ * HQ * CQ * PQ * NQ)   // 8,388,608 floats

typedef __attribute__((ext_vector_type(2))) float v2f;
typedef __attribute__((ext_vector_type(8))) float v8f;

__device__ __forceinline__ v8f wmma4(v2f a, v2f b, v8f c) {
  // D(16x16 f32) = A(16x4 f32) x B(4x16 f32) + C
  return __builtin_amdgcn_wmma_f32_16x16x4_f32(
      /*neg_a=*/false, a, /*neg_b=*/false, b,
      /*c_mod=*/(short)0, c, /*reuse_a=*/false, /*reuse_b=*/false);
}

// ---------------------------------------------------------------------------
// Kernel 1: per-(b,chunk,h) intra-chunk work.
//   G = C.B^T ; M = G o L (inclusive segsum mask) ; Y_diag = M.X -> d_out
//   S = (X * decay)^T . B -> wsStates[slot c] ; totA -> wsTot
// ---------------------------------------------------------------------------
__global__ __launch_bounds__(256) void ssd_chunk_kernel(
    const float* __restrict__ Xg, const float* __restrict__ Ag,
    const float* __restrict__ Bg, const float* __restrict__ Cg,
    float* __restrict__ Yg, float* __restrict__ wsStates,
    float* __restrict__ wsTot) {
  const int cc = blockIdx.x, h = blockIdx.y, bb = blockIdx.z;
  __shared__ float sC[BL * LS];
  __shared__ float sB[BL * LS];
  __shared__ float sX[BL * LS];
  __shared__ float sG[BL * LS];
  __shared__ float sAv[BL], sCum[BL], sDec[BL];

  const int tid  = threadIdx.x;
  const int lane = tid & 31, wv = tid >> 5;
  const int ln = lane & 15, hi = lane >> 4;

  // Wave's two output tiles: (tm0, tn) and (tm0+2, tn) -> shared B fragments.
  const int tm0 = wv >> 2, tn = wv & 3;
  const int arow0 = tm0 * 16 + ln;      // A-fragment row, tile 0
  const int arow1 = arow0 + 32;         // A-fragment row, tile 1
  const int bcol  = tn * 16 + ln;       // B-fragment column (shared)
  const int rb0 = tm0 * 16 + hi * 8;    // D rows, tile 0
  const int rb1 = rb0 + 32;             // D rows, tile 1

  // base flat index of row l=0 of this chunk in (b,s,h,p|n) layout
  const long rowbase0 = ((long)(bb * SQ + cc * BL) * HQ + h) * 64;

  // Stage X, B, C tiles (b128 global loads, b128 LDS stores; rows 16B aligned)
  for (int i = tid; i < BL * 16; i += 256) {
    const int l = i >> 4, j4 = (i & 15) * 4;
    const long g = rowbase0 + (long)l * (HQ * 64) + j4;
    *(float4*)(sX + l * LS + j4) = *(const float4*)(Xg + g);
    *(float4*)(sB + l * LS + j4) = *(const float4*)(Bg + g);
    *(float4*)(sC + l * LS + j4) = *(const float4*)(Cg + g);
  }
  if (tid < BL) sAv[tid] = Ag[(long)(bb * SQ + cc * BL + tid) * HQ + h];
  __syncthreads();

  if (tid < BL) {                       // inclusive cumsum of A over the chunk
    float s = 0.f;
    for (int i = 0; i <= tid; ++i) s += sAv[i];
    sCum[tid] = s;
  }
  __syncthreads();
  const float tot = sCum[BL - 1];
  if (tid < BL) sDec[tid] = __expf(tot - sCum[tid]);   // consumed after later barrier

  // ---- G = C . B^T  (k over n; both fragments k-contiguous float2) ----
  {
    v8f acc0 = {}, acc1 = {};
#pragma unroll
    for (int kk = 0; kk < 16; ++kk) {
      const int k0 = kk * 4 + hi * 2;
      v2f b  = *(const v2f*)(sB + bcol * LS + k0);
      v2f a0 = *(const v2f*)(sC + arow0 * LS + k0);
      v2f a1 = *(const v2f*)(sC + arow1 * LS + k0);
      acc0 = wmma4(a0, b, acc0);
      acc1 = wmma4(a1, b, acc1);
    }
#pragma unroll
    for (int v = 0; v < 8; ++v) {
      sG[(rb0 + v) * LS + bcol] = acc0[v];
      sG[(rb1 + v) * LS + bcol] = acc1[v];
    }
  }
  __syncthreads();

  // ---- apply inclusive-segsum decay mask: L[l,s] = exp(cum[l]-cum[s]+a[s]), l>=s
  for (int i = tid; i < BL * BL; i += 256) {
    const int l = i >> 6, s = i & 63;
    float g = sG[l * LS + s];
    sG[l * LS + s] = (l >= s) ? g * __expf(sCum[l] - sCum[s] + sAv[s]) : 0.f;
  }
  __syncthreads();

  // ---- Y_diag = M . X  (A k-contiguous from sG; B k strides rows of sX) ----
  {
    v8f acc0 = {}, acc1 = {};
#pragma unroll
    for (int kk = 0; kk < 16; ++kk) {
      const int k0 = kk * 4 + hi * 2;
      v2f b;
      b.x = sX[k0 * LS + bcol];
      b.y = sX[(k0 + 1) * LS + bcol];
      v2f a0 = *(const v2f*)(sG + arow0 * LS + k0);
      v2f a1 = *(const v2f*)(sG + arow1 * LS + k0);
      acc0 = wmma4(a0, b, acc0);
      acc1 = wmma4(a1, b, acc1);
    }
#pragma unroll
    for (int v = 0; v < 8; ++v) {
      Yg[rowbase0 + (long)(rb0 + v) * (HQ * 64) + bcol] = acc0[v];
      Yg[rowbase0 + (long)(rb1 + v) * (HQ * 64) + bcol] = acc1[v];
    }
  }
  __syncthreads();

  // ---- scale X rows by decay_states[l] = exp(tot - cum[l]) ----
  for (int i = tid; i < BL * BL; i += 256) {
    const int l = i >> 6, p = i & 63;
    sX[l * LS + p] *= sDec[l];
  }
  __syncthreads();

  // ---- S[p][n] = sum_l Xd[l][p] * B[l][n]  (k over l: strided reads) ----
  const long stBase = ((long)((bb * HQ + h) * CQ + cc)) * (PQ * NQ);
  {
    v8f acc0 = {}, acc1 = {};
#pragma unroll
    for (int kk = 0; kk < 16; ++kk) {
      const int k0 = kk * 4 + hi * 2;
      v2f b, a0, a1;
      b.x  = sB[k0 * LS + bcol];
      b.y  = sB[(k0 + 1) * LS + bcol];
      a0.x = sX[k0 * LS + arow0 - 0];         // column arow0 of X -> row p of Xd^T
      a0.y = sX[(k0 + 1) * LS + arow0];
      a1.x = sX[k0 * LS + arow1];
      a1.y = sX[(k0 + 1) * LS + arow1];
      acc0 = wmma4(a0, b, acc0);
      acc1 = wmma4(a1, b, acc1);
    }
#pragma unroll
    for (int v = 0; v < 8; ++v) {
      wsStates[stBase + (rb0 + v) * NQ + bcol] = acc0[v];
      wsStates[stBase + (rb1 + v) * NQ + bcol] = acc1[v];
    }
  }
  if (tid == 0) wsTot[(bb * HQ + h) * CQ + cc] = tot;
}

// ---------------------------------------------------------------------------
// Kernel 2: inter-chunk scan per (b,h): T[z] = exp(totA[z-1]) * (T[z-1] + S[z-1]).
// T[z] is written IN PLACE into slot z-1 (read happens first), so chunk z's
// entering state lives in slot z-1. T[0] == 0 (chunk 0 skipped in kernel 3).
// ---------------------------------------------------------------------------
__global__ __launch_bounds__(256) void ssd_scan_kernel(
    float* __restrict__ states, const float* __restrict__ tot) {
  const int bh = blockIdx.x;            // 0..31
  const int tid = threadIdx.x;
  __shared__ float sE[CQ];
  if (tid < CQ) sE[tid] = __expf(tot[bh * CQ + tid]);
  __syncthreads();

  float t[16];
#pragma unroll
  for (int r = 0; r < 16; ++r) t[r] = 0.f;

  const long base = (long)bh * CQ * (PQ * NQ);
  for (int z = 1; z < CQ; ++z) {
    const float e = sE[z - 1];
    const long slot = base + (long)(z - 1) * (PQ * NQ);
#pragma unroll
    for (int r = 0; r < 16; ++r) {
      const float s = states[slot + r * 256 + tid];
      t[r] = e * (t[r] + s);
    }
#pragma unroll
    for (int r = 0; r < 16; ++r) states[slot + r * 256 + tid] = t[r];
  }
}

// ---------------------------------------------------------------------------
// Kernel 3: Y += exp(cumA[l]) * (C . T^T)  for chunks 1..63.
// ---------------------------------------------------------------------------
__global__ __launch_bounds__(256) void ssd_off_kernel(
    const float* __restrict__ Ag, const float* __restrict__ Cg,
    const float* __restrict__ states, float* __restrict__ Yg) {
  const int cc = blockIdx.x + 1, h = blockIdx.y, bb = blockIdx.z;
  __shared__ float sC[BL * LS];
  __shared__ float sT[PQ * LS];
  __shared__ float sAv[BL], sEcum[BL];

  const int tid  = threadIdx.x;
  const int lane = tid & 31, wv = tid >> 5;
  const int ln = lane & 15, hi = lane >> 4;

  const int tm0 = wv >> 2, tn = wv & 3;
  const int arow0 = tm0 * 16 + ln, arow1 = arow0 + 32;
  const int bcol  = tn * 16 + ln;
  const int rb0 = tm0 * 16 + hi * 8, rb1 = rb0 + 32;

  const long rowbase0 = ((long)(bb * SQ + cc * BL) * HQ + h) * 64;
  const long stBase = ((long)((bb * HQ + h) * CQ + (cc - 1))) * (PQ * NQ);

  for (int i = tid; i < BL * 16; i += 256) {
    const int l = i >> 4, j4 = (i & 15) * 4;
    *(float4*)(sC + l * LS + j4) =
        *(const float4*)(Cg + rowbase0 + (long)l * (HQ * 64) + j4);
    *(float4*)(sT + l * LS + j4) =
        *(const float4*)(states + stBase + (long)l * NQ + j4);
  }
  if (tid < BL) sAv[tid] = Ag[(long)(bb * SQ + cc * BL + tid) * HQ + h];
  __syncthreads();
  if (tid < BL) {
    float s = 0.f;
    for (int i = 0; i <= tid; ++i) s += sAv[i];
    sEcum[tid] = __expf(s);              // state_decay_out[l]
  }
  __syncthreads();

  // Y_off[l][p] = sEcum[l] * sum_n C[l][n] * T[p][n]  (both k-contiguous)
  v8f acc0 = {}, acc1 = {};
#pragma unroll
  for (int kk = 0; kk < 16; ++kk) {
    const int k0 = kk * 4 + hi * 2;
    v2f b  = *(const v2f*)(sT + bcol * LS + k0);
    v2f a0 = *(const v2f*)(sC + arow0 * LS + k0);
    v2f a1 = *(const v2f*)(sC + arow1 * LS + k0);
    acc0 = wmma4(a0, b, acc0);
    acc1 = wmma4(a1, b, acc1);
  }
#pragma unroll
  for (int v = 0; v < 8; ++v) {
    const long o0 = rowbase0 + (long)(rb0 + v) * (HQ * 64) + bcol;
    const long o1 = rowbase0 + (long)(rb1 + v) * (HQ * 64) + bcol;
    Yg[o0] += sEcum[rb0 + v] * acc0[v];
    Yg[o1] += sEcum[rb1 + v] * acc1[v];
  }
}

extern "C" void kernel_launch(void* const* d_in, const int* in_sizes, int n_in,
                              void* d_out, int out_size, void* d_ws, size_t ws_size,
                              hipStream_t stream) {
  (void)in_sizes; (void)n_in; (void)out_size; (void)ws_size;
  const float* X = (const float*)d_in[0];
  const float* A = (const float*)d_in[1];
  const float* B = (const float*)d_in[2];
  const float* C = (const float*)d_in[3];
  float* Y = (float*)d_out;
  float* wsStates = (float*)d_ws;                 // 32 MB
  float* wsTot    = wsStates + STATES_ELEMS;      // 2048 floats

  dim3 blk(256);
  ssd_chunk_kernel<<<dim3(CQ, HQ, BQ), blk, 0, stream>>>(X, A, B, C, Y,
                                                         wsStates, wsTot);
  ssd_scan_kernel<<<dim3(BQ * HQ), blk, 0, stream>>>(wsStates, wsTot);
  ssd_off_kernel<<<dim3(CQ - 1, HQ, BQ), blk, 0, stream>>>(A, C, wsStates, Y);
}